// TreeCompositionGRU_55508157333689
// MI455X (gfx1250) — compile-verified
//
#include <hip/hip_runtime.h>

typedef __attribute__((ext_vector_type(2))) float v2f;
typedef __attribute__((ext_vector_type(8))) float v8f;
typedef __attribute__((address_space(3))) float lds_f;

#define WORD_DIM 300
#define TAG_DIM  50
#define REL_DIM  50
#define MEM_DIM  300
#define IN_DIM   700            // 300+50+50+300
#define GATE_DIM 900            // 3*MEM_DIM
#define BATCH    1024
#define TSTEPS   64
#define MROWS    (BATCH * TSTEPS)   // 65536
#define HP_LD    912            // padded leading dim for per-step hp buffer

#define KC       100            // K chunk (divides 700 and 300); 25 k-steps of 4
#define KSTEPS   (KC / 4)       // 25
#define PAD_LD   108            // LDS floats per column: 16B-aligned stride, 108%64=44 -> conflict-free
#define NSTRIP   64             // output columns per workgroup (4 WMMA tiles)
#define MT_PER_WG 8             // one 16-row M tile per wave, 8 waves per WG

// ---------------------------------------------------------------------------
// Pack concatenated input x = [words | tags | rels | child] into [MROWS, 700]
// ---------------------------------------------------------------------------
__global__ void pack_x_kernel(const float* __restrict__ words,
                              const float* __restrict__ tags,
                              const float* __restrict__ rels,
                              const float* __restrict__ child,
                              float* __restrict__ xpack) {
  long idx = (long)blockIdx.x * blockDim.x + threadIdx.x;
  const long total = (long)MROWS * IN_DIM;
  if (idx >= total) return;
  long m = idx / IN_DIM;
  int  k = (int)(idx - m * IN_DIM);
  float v;
  if (k < WORD_DIM) {
    v = words[m * WORD_DIM + k];
  } else if (k < WORD_DIM + TAG_DIM) {
    v = tags[m * TAG_DIM + (k - WORD_DIM)];
  } else if (k < WORD_DIM + TAG_DIM + REL_DIM) {
    v = rels[m * REL_DIM + (k - WORD_DIM - TAG_DIM)];
  } else {
    v = child[m * MEM_DIM + (k - WORD_DIM - TAG_DIM - REL_DIM)];
  }
  xpack[idx] = v;
}

__global__ void zero_kernel(float* __restrict__ p, int n) {
  int idx = blockIdx.x * blockDim.x + threadIdx.x;
  if (idx < n) p[idx] = 0.0f;
}

// ---------------------------------------------------------------------------
// C[M,N] = A[M,K] * W[N,K]^T with V_WMMA_F32_16X16X4_F32.
//
// Workgroup (8 waves) owns a 128(M) x 64(N) output block.
// B (weights) are staged in LDS with GLOBAL_LOAD_ASYNC_TO_LDS_B128,
// double-buffered per 100-wide K chunk (s_wait_asynccnt + barrier).
// Each wave prefetches its whole A chunk (25 x v2f) into registers, then
// runs 25 k-steps x 4 WMMAs reading B fragments from LDS (ds_load_b64).
//
// Fragment layouts per ISA 7.12.2 (f32, wave32):
//   A 16x4: lane<16 holds {A[r][k0],A[r][k0+1]}, lane>=16 {A[r][k0+2],A[r][k0+3]}
//   B 4x16 mirrors with rows<->cols swapped.
//   D 16x16: vgpr rr, lane l -> row rr + 8*(l>>4), col l&15.
// ---------------------------------------------------------------------------
__global__ void __launch_bounds__(256)
wmma_gemm_nt_kernel(const float* __restrict__ A, int lda,
                    const float* __restrict__ W, int K,
                    float* __restrict__ C, int ldc,
                    int N, int Ntiles, int NStrips) {
  __shared__ float Bsh[2][NSTRIP * PAD_LD];

  const int strip  = blockIdx.x % NStrips;
  const int mchunk = blockIdx.x / NStrips;
  const int tid    = threadIdx.x;
  const int wv     = tid >> 5;
  const int lane   = tid & 31;
  const int half   = lane >> 4;   // 0: K+0..1, 1: K+2..3
  const int r      = lane & 15;

  const int mtile   = mchunk * MT_PER_WG + wv;
  const int nchunks = K / KC;

  const float* Arow = A + (long)(mtile * 16 + r) * lda + 2 * half;

  v8f acc[4] = {};

  // Cooperatively issue async copies of one 64-col x KC-k weight chunk to LDS.
  auto issue_chunk = [&](int c, int buf) {
    const int base_col = strip * NSTRIP;
    for (int i = tid; i < NSTRIP * KSTEPS; i += 256) {   // 1600 x b128
      int col = i / KSTEPS;
      int off = i - col * KSTEPS;                        // b128 index in col
      int gcol = base_col + col;
      if (gcol >= N) gcol = N - 1;                       // clamp: harmless dup
      const float* g = W + (long)gcol * K + c * KC + off * 4;   // 16B aligned
      lds_f* l = (lds_f*)&Bsh[buf][col * PAD_LD + off * 4];     // 16B aligned
      asm volatile("global_load_async_to_lds_b128 %0, %1, off"
                   :: "v"(l), "v"(g) : "memory");
    }
  };

  issue_chunk(0, 0);
  asm volatile("s_wait_asynccnt 0x0" ::: "memory");
  __syncthreads();

  for (int c = 0; c < nchunks; ++c) {
    const int buf = c & 1;
    if (c + 1 < nchunks) issue_chunk(c + 1, buf ^ 1);

    // Batch-prefetch this wave's A chunk into registers (one wait for all).
    v2f areg[KSTEPS];
#pragma unroll
    for (int s = 0; s < KSTEPS; ++s)
      areg[s] = *(const v2f*)(Arow + c * KC + 4 * s);

    const float* Bbase = &Bsh[buf][0];
#pragma unroll
    for (int s = 0; s < KSTEPS; ++s) {
      const int kk = 4 * s + 2 * half;
#pragma unroll
      for (int j = 0; j < 4; ++j) {
        const int cl = j * 16 + r;                       // strip-local column
        v2f b = *(const v2f*)(Bbase + cl * PAD_LD + kk); // ds_load_b64
        acc[j] = __builtin_amdgcn_wmma_f32_16x16x4_f32(
            /*neg_a=*/false, areg[s], /*neg_b=*/false, b,
            /*c_mod=*/(short)0, acc[j], /*reuse_a=*/false, /*reuse_b=*/false);
      }
    }

    asm volatile("s_wait_asynccnt 0x0" ::: "memory");
    __syncthreads();
  }

#pragma unroll
  for (int j = 0; j < 4; ++j) {
    const int ntile = strip * 4 + j;
    if (ntile >= Ntiles) break;
    const int col = ntile * 16 + r;
    if (col < N) {
#pragma unroll
      for (int rr = 0; rr < 8; ++rr) {
        const int row = mtile * 16 + rr + 8 * half;
        C[(long)row * ldc + col] = acc[j][rr];
      }
    }
  }
}

// ---------------------------------------------------------------------------
// GRU gate math for one time step.
// ---------------------------------------------------------------------------
__global__ void gru_gate_kernel(const float* __restrict__ xproj,  // [MROWS,900]
                                const float* __restrict__ hp,     // [BATCH,HP_LD]
                                const float* __restrict__ hprev,  // [BATCH,300]
                                const float* __restrict__ b_ih,   // [900]
                                const float* __restrict__ b_hh,   // [900]
                                float* __restrict__ hnew,         // [BATCH,300]
                                float* __restrict__ out,          // [BATCH,T,300]
                                int t) {
  const int idx = blockIdx.x * blockDim.x + threadIdx.x;
  if (idx >= BATCH * MEM_DIM) return;
  const int b = idx / MEM_DIM;
  const int j = idx - b * MEM_DIM;

  const long xrow = ((long)b * TSTEPS + t) * GATE_DIM;
  const float* hpr = hp + (long)b * HP_LD;

  const float xr = xproj[xrow + j]             + b_ih[j];
  const float xz = xproj[xrow + MEM_DIM + j]   + b_ih[MEM_DIM + j];
  const float xn = xproj[xrow + 2*MEM_DIM + j] + b_ih[2*MEM_DIM + j];
  const float hr = hpr[j]             + b_hh[j];
  const float hz = hpr[MEM_DIM + j]   + b_hh[MEM_DIM + j];
  const float hn = hpr[2*MEM_DIM + j] + b_hh[2*MEM_DIM + j];

  const float rg = 1.0f / (1.0f + __expf(-(xr + hr)));
  const float zg = 1.0f / (1.0f + __expf(-(xz + hz)));
  const float ng = tanhf(xn + rg * hn);
  const float h  = hprev[idx];
  const float hv = (1.0f - zg) * ng + zg * h;

  hnew[idx] = hv;
  out[((long)b * TSTEPS + t) * MEM_DIM + j] = hv;
}

// ---------------------------------------------------------------------------
extern "C" void kernel_launch(void* const* d_in, const int* in_sizes, int n_in,
                              void* d_out, int out_size, void* d_ws, size_t ws_size,
                              hipStream_t stream) {
  (void)in_sizes; (void)n_in; (void)out_size; (void)ws_size;

  const float* words = (const float*)d_in[0];
  const float* tags  = (const float*)d_in[1];
  const float* rels  = (const float*)d_in[2];
  const float* child = (const float*)d_in[3];
  const float* w_ih  = (const float*)d_in[4];   // [900,700]
  const float* w_hh  = (const float*)d_in[5];   // [900,300]
  const float* b_ih  = (const float*)d_in[6];   // [900]
  const float* b_hh  = (const float*)d_in[7];   // [900]
  float* out = (float*)d_out;                   // [B,T,300]

  // Workspace carve-up (floats)
  float* xpack = (float*)d_ws;                              // MROWS*IN_DIM
  float* xproj = xpack + (size_t)MROWS * IN_DIM;            // MROWS*GATE_DIM
  float* hpbuf = xproj + (size_t)MROWS * GATE_DIM;          // BATCH*HP_LD
  float* h0    = hpbuf + (size_t)BATCH * HP_LD;             // BATCH*MEM_DIM
  float* h1    = h0    + (size_t)BATCH * MEM_DIM;           // BATCH*MEM_DIM

  const int Ntiles  = (GATE_DIM + 15) / 16;  // 57
  const int NStrips = (Ntiles + 3) / 4;      // 15

  // 1) pack concatenated inputs
  {
    const long total = (long)MROWS * IN_DIM;
    pack_x_kernel<<<(unsigned)((total + 255) / 256), 256, 0, stream>>>(
        words, tags, rels, child, xpack);
  }
  // 2) h0 = 0
  zero_kernel<<<(BATCH * MEM_DIM + 255) / 256, 256, 0, stream>>>(h0, BATCH * MEM_DIM);

  // 3) big projection GEMM: xproj = xpack @ w_ih^T   (M=65536,K=700,N=900)
  {
    const int Mchunks = (MROWS / 16) / MT_PER_WG;  // 512
    wmma_gemm_nt_kernel<<<Mchunks * NStrips, 256, 0, stream>>>(
        xpack, IN_DIM, w_ih, IN_DIM, xproj, GATE_DIM, GATE_DIM, Ntiles, NStrips);
  }

  // 4) recurrent scan: 64 x (GEMM + gate)
  float* hcur = h0;
  float* hnxt = h1;
  for (int t = 0; t < TSTEPS; ++t) {
    const int Mchunks = (BATCH / 16) / MT_PER_WG;  // 8
    wmma_gemm_nt_kernel<<<Mchunks * NStrips, 256, 0, stream>>>(
        hcur, MEM_DIM, w_hh, MEM_DIM, hpbuf, HP_LD, GATE_DIM, Ntiles, NStrips);
    gru_gate_kernel<<<(BATCH * MEM_DIM + 255) / 256, 256, 0, stream>>>(
        xproj, hpbuf, hcur, b_ih, b_hh, hnxt, out, t);
    float* tmp = hcur; hcur = hnxt; hnxt = tmp;
  }
}